// LMNNLoss_SP_OPT_7146825581135
// MI455X (gfx1250) — compile-verified
//
#include <hip/hip_runtime.h>

// LMNN loss, specialized: n=128, p=512, d=64, K_TARGETS=15, N_CLASSES=8.
//
// Roofline: dominated by a single pass over `outputs` (16.8 MB @ 23.3 TB/s
// ~0.7us). Plan:
//   K1: dist_c[i,j] = sum_d (out - center)^2, d-reduction done EXACTLY in f32
//       via V_WMMA_F32_16X16X4_F32 with B = ones (every D column = row sum).
//   K2: per-row (i) label bookkeeping: per class c, S/F over the first 15
//       same-class indices (stable top_k of constant values == first-15),
//       class counts, then pull/push partial per row (no global fp atomics).
//   K3: reduce 128 row partials, scale by 1/(n*p).

#define LN 128
#define LP 512
#define LD 64
#define LK 15
#define LC 8

typedef __attribute__((ext_vector_type(2))) float v2f;
typedef __attribute__((ext_vector_type(8))) float v8f;

// ---------------- Kernel 1: dist_c via WMMA f32 16x16x4 ----------------
// One wave32 per 16-wide j tile. A-matrix layout (32-bit A 16x4):
//   lanes 0-15 : M=lane,    VGPR0=K0, VGPR1=K1
//   lanes 16-31: M=lane-16, VGPR0=K2, VGPR1=K3
// D-matrix (32-bit 16x16): VGPR r holds M=r (lanes 0-15) / M=r+8 (lanes 16-31).
__global__ __launch_bounds__(128) void lmnn_dist_kernel(
    const float* __restrict__ center,   // [LN, LD]
    const float* __restrict__ outputs,  // [LN, LP, LD]
    float* __restrict__ dist)           // [LN, LP]
{
    const int wave  = threadIdx.x >> 5;
    const int lane  = threadIdx.x & 31;
    const int tile  = blockIdx.x * 4 + wave;   // LN * LP/16 = 4096 tiles total
    const int i     = tile >> 5;               // LP/16 = 32 tiles per row
    const int jb    = (tile & 31) << 4;
    const int m     = lane & 15;
    const int khalf = (lane >> 4) << 1;        // 0 for lanes 0-15, 2 for 16-31

    const float* orow = outputs + ((size_t)i * LP + jb + m) * LD;
    const float* crow = center + (size_t)i * LD;

#if __has_builtin(__builtin_amdgcn_wmma_f32_16x16x4_f32)
    v8f acc = {};
    const v2f ones = {1.0f, 1.0f};
#pragma unroll
    for (int t = 0; t < LD / 4; ++t) {
        const int k = t * 4 + khalf;
        v2f ov = *(const v2f*)(orow + k);
        v2f cv = *(const v2f*)(crow + k);
        v2f df = ov - cv;
        v2f sq = df * df;
        // D = SQ(16x4) * ones(4x16) + C  -> every column n of D is the
        // partial row-sum; accumulate over 16 chunks of K=4 (exact f32).
        acc = __builtin_amdgcn_wmma_f32_16x16x4_f32(
            /*neg_a=*/false, sq, /*neg_b=*/false, ones,
            /*c_mod=*/(short)0, acc, /*reuse_a=*/false, /*reuse_b=*/false);
    }
    float* drow = dist + (size_t)i * LP + jb;
    if (lane == 0) {
        *(v8f*)(drow) = acc;          // M=0..7
    } else if (lane == 16) {
        *(v8f*)(drow + 8) = acc;      // M=8..15
    }
#else
    // VALU fallback: lane L sums its K half, partner lane (L^16) has the rest.
    float part = 0.0f;
#pragma unroll
    for (int t = 0; t < LD / 4; ++t) {
        const int k = t * 4 + khalf;
        v2f ov = *(const v2f*)(orow + k);
        v2f cv = *(const v2f*)(crow + k);
        v2f df = ov - cv;
        part += df.x * df.x + df.y * df.y;
    }
    float sum = part + __shfl_xor(part, 16, 32);
    if (lane < 16) dist[(size_t)i * LP + jb + m] = sum;
#endif
}

// ---------------- Kernel 2: per-row label logic ----------------
__global__ __launch_bounds__(256) void lmnn_row_kernel(
    const int* __restrict__ labels,   // [LN, LP]
    const float* __restrict__ dist,   // [LN, LP]
    float* __restrict__ partials)     // [LN]
{
    __shared__ float sd[LP];
    __shared__ int   sl[LP];
    __shared__ int   scnt[LC];
    __shared__ float sS[LC];
    __shared__ float sF[LC];
    __shared__ float sred[256];

    const int i   = blockIdx.x;
    const int tid = threadIdx.x;

    for (int j = tid; j < LP; j += 256) {
        sd[j] = dist[(size_t)i * LP + j];
        sl[j] = labels[(size_t)i * LP + j];
    }
    if (tid < LC) scnt[tid] = 0;
    __syncthreads();

    for (int j = tid; j < LP; j += 256) atomicAdd(&scnt[sl[j]], 1);
    __syncthreads();

    // First-15 per class, in index order (matches stable top_k on ties).
    if (tid < LC) {
        const int c = tid;
        int taken = 0;
        float S = 0.0f, F = -__builtin_inff();
        for (int j = 0; j < LP && taken < LK; ++j) {
            if (sl[j] == c) {
                S += sd[j];
                F = fmaxf(F, sd[j]);
                ++taken;
            }
        }
        sS[c] = S;
        sF[c] = F;
    }
    __syncthreads();

    float mseg = -__builtin_inff();
#pragma unroll
    for (int c = 0; c < LC; ++c) mseg = fmaxf(mseg, sF[c]);
    mseg += 1.0f;  // margin_seg[i]

    float acc = 0.0f;
    for (int j = tid; j < LP; j += 256) {
        const int   c  = sl[j];
        const float dc = sd[j];
        acc += sS[c];  // pull contribution
        const float hinge = fmaxf(1.0f + sF[c] - dc, 0.0f);
        if (dc < mseg) acc += (float)(LP - scnt[c]) * hinge;  // push
    }
    sred[tid] = acc;
    __syncthreads();
    for (int s = 128; s > 0; s >>= 1) {
        if (tid < s) sred[tid] += sred[tid + s];
        __syncthreads();
    }
    if (tid == 0) partials[i] = sred[0];
}

// ---------------- Kernel 3: final reduce + scale ----------------
__global__ __launch_bounds__(128) void lmnn_final_kernel(
    const float* __restrict__ partials, float* __restrict__ out)
{
    __shared__ float sred[128];
    const int tid = threadIdx.x;
    sred[tid] = partials[tid];
    __syncthreads();
    for (int s = 64; s > 0; s >>= 1) {
        if (tid < s) sred[tid] += sred[tid + s];
        __syncthreads();
    }
    if (tid == 0) out[0] = sred[0] / (float)(LN * LP);
}

extern "C" void kernel_launch(void* const* d_in, const int* in_sizes, int n_in,
                              void* d_out, int out_size, void* d_ws, size_t ws_size,
                              hipStream_t stream) {
    const float* center  = (const float*)d_in[0];   // (128, 64)  f32
    const float* outputs = (const float*)d_in[1];   // (128, 512, 64) f32
    const int*   labels  = (const int*)d_in[2];     // (128, 512) i32
    float* out = (float*)d_out;

    float* dist     = (float*)d_ws;                 // LN*LP floats (256 KB)
    float* partials = dist + (size_t)LN * LP;       // LN floats

    // K1: 4096 wave-tiles, 4 waves (128 thr) per block -> 1024 blocks.
    lmnn_dist_kernel<<<(LN * (LP / 16)) / 4, 128, 0, stream>>>(center, outputs, dist);
    // K2: one block per row.
    lmnn_row_kernel<<<LN, 256, 0, stream>>>(labels, dist, partials);
    // K3: single block.
    lmnn_final_kernel<<<1, 128, 0, stream>>>(partials, out);
}